// BusEmbedding_20873541059064
// MI455X (gfx1250) — compile-verified
//
#include <hip/hip_runtime.h>
#include <math.h>

typedef __attribute__((ext_vector_type(2))) float v2f;
typedef __attribute__((ext_vector_type(8))) float v8f;

#define NROWS 1000000
#define DIM   128

// Branchless tanh on the hardware transcendental pipe:
//   tanh(x) = 1 - 2 / (exp(2x) + 1) = 1 - 2*rcp(exp2(x * 2*log2(e)) + 1)
// Limits work without clamping: x->+inf => 1, x->-inf => -1, NaN propagates.
__device__ __forceinline__ float fast_tanh(float x) {
    const float TWO_LOG2E = 2.88539008177792681472f; // 2*log2(e)
    float e = __builtin_amdgcn_exp2f(x * TWO_LOG2E); // v_exp_f32
    float r = __builtin_amdgcn_rcpf(e + 1.0f);       // v_rcp_f32
    return __builtin_fmaf(-2.0f, r, 1.0f);           // v_fma_f32
}

// One wave32 handles a 16-row tile of the output (16 x 128).
// GEMV (16x2)@(2x128) runs as 8 column-tiles of V_WMMA_F32_16X16X4_F32
// (K padded 2->4 with zeros), 3 weight sets per tile, bias in the C operand.
// Per-row select happens pre-activation (tanh is monotone elementwise,
// tanh(0)=0 matches the bt-not-in-{1,2,3} case), then tanh, then store.
__global__ __launch_bounds__(128) void BusEmbedding_kernel(
    const float* __restrict__ feat,
    const int*   __restrict__ bus_type,
    const float* __restrict__ Ws, const float* __restrict__ bs,
    const float* __restrict__ Wg, const float* __restrict__ bg,
    const float* __restrict__ Wl, const float* __restrict__ bl,
    float* __restrict__ out)
{
    const int lane = threadIdx.x & 31;
    const int wave = threadIdx.x >> 5;
    const int tile = blockIdx.x * 4 + wave;   // 16-row tile index
    const int rb   = tile * 16;               // first row of this tile

    // ---- A matrix (16x4 f32): lanes 0-15 carry K=0,1 for row M=lane;
    //      lanes 16-31 carry K=2,3 which are the zero padding.
    v2f a = {0.0f, 0.0f};
    if (lane < 16) {
        const float2 f = *(const float2*)(feat + (size_t)(rb + lane) * 2);
        a.x = f.x;  // K=0
        a.y = f.y;  // K=1
    }
    // (wave re-converges here; EXEC all-1s again before any WMMA)

    // D-matrix layout: VGPR r holds row rb+r (lanes 0-15) or rb+8+r (lanes 16-31).
    const int row0 = rb + ((lane >> 4) << 3);
    int btv[8];
    #pragma unroll
    for (int r = 0; r < 8; ++r) btv[r] = bus_type[row0 + r];

    const int  nlo = lane & 15;
    const bool lo  = (lane < 16);

    #pragma unroll
    for (int t = 0; t < 8; ++t) {
        const int col = t * 16 + nlo;

        // ---- B matrices (4x16 f32): lanes 0-15 carry K=0,1 for col N=lane;
        //      lanes 16-31 (K=2,3) are zero padding.  W is row-major (2,128).
        v2f Bs = {0.0f, 0.0f}, Bg = {0.0f, 0.0f}, Bl = {0.0f, 0.0f};
        if (lo) {
            Bs.x = Ws[col]; Bs.y = Ws[DIM + col];
            Bg.x = Wg[col]; Bg.y = Wg[DIM + col];
            Bl.x = Wl[col]; Bl.y = Wl[DIM + col];
        }

        // ---- C = bias broadcast down the 16 rows of this column tile.
        const float cs = bs[col], cg = bg[col], cl = bl[col];
        v8f accS, accG, accL;
        #pragma unroll
        for (int r = 0; r < 8; ++r) { accS[r] = cs; accG[r] = cg; accL[r] = cl; }

        // D = A x B + C on the matrix pipe (compute is free; store-BW bound).
        accS = __builtin_amdgcn_wmma_f32_16x16x4_f32(
                   false, a, false, Bs, (short)0, accS, false, false);
        accG = __builtin_amdgcn_wmma_f32_16x16x4_f32(
                   false, a, false, Bg, (short)0, accG, false, false);
        accL = __builtin_amdgcn_wmma_f32_16x16x4_f32(
                   false, a, false, Bl, (short)0, accL, false, false);

        // ---- per-row branchless select, tanh, store.
        // Each dword store covers rows {rb+r, rb+8+r}, cols t*16..t*16+15:
        // two contiguous 64B segments -> coalesces cleanly.
        float* op = out + (size_t)row0 * DIM + col;
        #pragma unroll
        for (int r = 0; r < 8; ++r) {
            const int b = btv[r];
            // independent selects -> v_cndmask chain, no exec-mask branching
            float pre = (b == 1) ? accS[r] : 0.0f;
            pre       = (b == 2) ? accG[r] : pre;
            pre       = (b == 3) ? accL[r] : pre;
            op[(size_t)r * DIM] = fast_tanh(pre);
        }
    }
}

extern "C" void kernel_launch(void* const* d_in, const int* in_sizes, int n_in,
                              void* d_out, int out_size, void* d_ws, size_t ws_size,
                              hipStream_t stream) {
    const float* feat     = (const float*)d_in[0];
    const int*   bus_type = (const int*)  d_in[1];
    const float* W_slack  = (const float*)d_in[2];
    const float* b_slack  = (const float*)d_in[3];
    const float* W_gen    = (const float*)d_in[4];
    const float* b_gen    = (const float*)d_in[5];
    const float* W_load   = (const float*)d_in[6];
    const float* b_load   = (const float*)d_in[7];
    float* out = (float*)d_out;

    // 128 threads = 4 waves; each wave owns 16 rows => 64 rows per block.
    // 1,000,000 / 64 = 15625 blocks exactly (no tail).
    const int blocks = NROWS / 64;
    BusEmbedding_kernel<<<blocks, 128, 0, stream>>>(
        feat, bus_type, W_slack, b_slack, W_gen, b_gen, W_load, b_load, out);
}